// SwinBlock_3272765079582
// MI455X (gfx1250) — compile-verified
//
#include <hip/hip_runtime.h>

typedef float v2f __attribute__((ext_vector_type(2)));
typedef float v4f __attribute__((ext_vector_type(4)));
typedef float v8f __attribute__((ext_vector_type(8)));
typedef int gv4i __attribute__((vector_size(16)));   // matches builtin param type

#define HW   50176   // 224*224
#define IMGW 224

#define AS1 __attribute__((address_space(1)))
#define AS3 __attribute__((address_space(3)))

#if defined(__HIP_DEVICE_COMPILE__) &&                                     \
    __has_builtin(__builtin_amdgcn_global_load_async_to_lds_b128) &&       \
    __has_builtin(__builtin_amdgcn_global_load_async_to_lds_b32) &&        \
    __has_builtin(__builtin_amdgcn_s_wait_asynccnt)
#define HAVE_ASYNC_LDS 1
#else
#define HAVE_ASYNC_LDS 0
#endif

__device__ __forceinline__ void async_b128(const float* g, float* l) {
#if HAVE_ASYNC_LDS
  __builtin_amdgcn_global_load_async_to_lds_b128((AS1 gv4i*)g, (AS3 gv4i*)l,
                                                 0, 0);
#else
  (void)g; (void)l;
#endif
}
__device__ __forceinline__ void async_b32(const float* g, float* l) {
#if HAVE_ASYNC_LDS
  __builtin_amdgcn_global_load_async_to_lds_b32((AS1 int*)g, (AS3 int*)l,
                                                0, 0);
#else
  (void)g; (void)l;
#endif
}
__device__ __forceinline__ void async_wait() {
#if HAVE_ASYNC_LDS
  __builtin_amdgcn_s_wait_asynccnt(0);
#endif
}

__device__ __forceinline__ v8f wmma_f32(v2f a, v2f b, v8f c) {
  // D = A(16x4) * B(4x16) + C, fp32 on the CDNA5 matrix pipe
  return __builtin_amdgcn_wmma_f32_16x16x4_f32(false, a, false, b, (short)0, c,
                                               false, false);
}
__device__ __forceinline__ v8f vzero() {
  v8f z = {0.f, 0.f, 0.f, 0.f, 0.f, 0.f, 0.f, 0.f};
  return z;
}
// original-coordinate pixel for shifted-window (wy,wx), element n = r*4+c
__device__ __forceinline__ int win_pix(int wy, int wx, int n) {
  int r = n >> 2, c = n & 3;
  int oy = wy * 4 + r + 2; if (oy >= IMGW) oy -= IMGW;
  int ox = wx * 4 + c + 2; if (ox >= IMGW) ox -= IMGW;
  return oy * IMGW + ox;
}
// Swin shift-mask region label of a shifted coordinate
__device__ __forceinline__ int region3(int s) {
  return (s < IMGW - 4) ? 0 : ((s < IMGW - 2) ? 1 : 2);
}

// ---------------- kernel 1: per (b,c) plane mean ----------------
__global__ void __launch_bounds__(256) k_mean(const float* __restrict__ x,
                                              float* __restrict__ g0) {
  const int blk = blockIdx.x;              // b*128 + c
  const float* p = x + (long)blk * HW;
  float s = 0.f;
  for (int i = threadIdx.x; i < HW; i += 256)
    s += __builtin_nontemporal_load(p + i);         // keep x out of L2
  __shared__ float r[256];
  r[threadIdx.x] = s;
  __syncthreads();
  for (int st = 128; st > 0; st >>= 1) {
    if (threadIdx.x < st) r[threadIdx.x] += r[threadIdx.x + st];
    __syncthreads();
  }
  if (threadIdx.x == 0) g0[blk] = r[0] * (1.f / (float)HW);
}

// ---------------- kernel 2: gating MLP 128->64->128 ----------------
__global__ void __launch_bounds__(128) k_gate(const float* __restrict__ g0,
                                              const float* __restrict__ w1,
                                              const float* __restrict__ b1,
                                              const float* __restrict__ w2,
                                              const float* __restrict__ b2,
                                              float* __restrict__ g) {
  const int b = blockIdx.x, t = threadIdx.x;
  __shared__ float gin[128];
  __shared__ float hid[64];
  gin[t] = g0[b * 128 + t];
  __syncthreads();
  if (t < 64) {
    float s = b1[t];
    for (int c = 0; c < 128; ++c) s += gin[c] * w1[t * 128 + c];
    hid[t] = fmaxf(s, 0.f);
  }
  __syncthreads();
  float s = b2[t];
  for (int k = 0; k < 64; ++k) s += hid[k] * w2[t * 64 + k];
  g[b * 128 + t] = s;
}

// ---------------- kernel 3: proj_in (128->96) GEMM, NCHW -> (row,96) -------
__global__ void __launch_bounds__(128) k_proj_in(const float* __restrict__ x,
                                                 const float* __restrict__ w,
                                                 const float* __restrict__ bias,
                                                 float* __restrict__ feat) {
  extern __shared__ float lds[];           // wT[c][o] : 128*96 floats
  const int tid = threadIdx.x;
#if HAVE_ASYNC_LDS
  for (int i = tid; i < 128 * 96; i += 128) {
    int c = i / 96, o = i - c * 96;
    async_b32(w + o * 128 + c, lds + i);   // DMA transpose gather to LDS
  }
  async_wait();
#else
  for (int i = tid; i < 128 * 96; i += 128) {
    int c = i / 96, o = i - c * 96;
    lds[i] = w[o * 128 + c];
  }
#endif
  __syncthreads();
  const int wave = tid >> 5, lane = tid & 31;
  const long row0 = ((long)blockIdx.x * 4 + wave) * 16;   // b*HW + p
  const int b  = (int)(row0 / HW);
  const int p0 = (int)(row0 - (long)b * HW);
  const float* xb = x + (long)b * 128 * HW;
  const int mrow = lane & 15;
  const int koff = (lane < 16) ? 0 : 2;
  const int prow = (lane < 16) ? 0 : 8;
  v8f acc[6];
#pragma unroll
  for (int nt = 0; nt < 6; ++nt) acc[nt] = vzero();
  for (int kt = 0; kt < 32; ++kt) {
    const int k = kt * 4 + koff;
    v2f a;
    a.x = __builtin_nontemporal_load(xb + (long)k * HW + p0 + mrow);
    a.y = __builtin_nontemporal_load(xb + (long)(k + 1) * HW + p0 + mrow);
    if (kt < 31)
      __builtin_prefetch(xb + (long)(k + 4) * HW + p0 + mrow, 0, 3);
#pragma unroll
    for (int nt = 0; nt < 6; ++nt) {
      v2f bf;
      bf.x = lds[k * 96 + nt * 16 + mrow];
      bf.y = lds[(k + 1) * 96 + nt * 16 + mrow];
      acc[nt] = wmma_f32(a, bf, acc[nt]);
    }
  }
  float* dst = feat + row0 * 96;
#pragma unroll
  for (int nt = 0; nt < 6; ++nt) {
    const float bv = bias[nt * 16 + mrow];
#pragma unroll
    for (int gi = 0; gi < 8; ++gi)
      dst[(gi + prow) * 96 + nt * 16 + mrow] = acc[nt][gi] + bv;
  }
}

// ---- kernel 4: fused LN + shifted-window MHA + proj + residual (in place) --
__global__ void __launch_bounds__(128) k_attn(float* __restrict__ feat,
                                              const float* __restrict__ n1w,
                                              const float* __restrict__ n1b,
                                              const float* __restrict__ qkvw,
                                              const float* __restrict__ qkvb,
                                              const float* __restrict__ relb,
                                              const float* __restrict__ apw,
                                              const float* __restrict__ apb) {
  extern __shared__ float lds[];
  const int tid = threadIdx.x, wave = tid >> 5, lane = tid & 31;
  float* Xraw = lds + wave * 7936;   // 16x96 raw (shortcut)
  float* Xln  = Xraw + 1536;         // 16x96 normed, reused as O
  float* QKV  = Xraw + 3072;         // 16x288
  float* Pm   = Xraw + 7680;         // 16x16 scores / probs

  const int wid = blockIdx.x * 4 + wave;     // window id in [0, 8*3136)
  const int b  = wid / 3136;
  const int w  = wid - b * 3136;
  const int wy = w / 56;
  const int wx = w - wy * 56;
  const long bbase = (long)b * HW * 96;

#if HAVE_ASYNC_LDS
  // async DMA the 16x96 window tile (24 x b128 per row) straight to LDS
  for (int i = lane; i < 384; i += 32) {
    int n = i / 24, q = i - n * 24;
    async_b128(feat + bbase + (long)win_pix(wy, wx, n) * 96 + q * 4,
               Xraw + i * 4);
  }
  async_wait();
#else
  for (int i = lane; i < 1536; i += 32) {
    int n = i / 96, c = i - n * 96;
    Xraw[i] = feat[bbase + (long)win_pix(wy, wx, n) * 96 + c];
  }
#endif
  if (lane < 16) {                    // LayerNorm, one row per lane
    const float* xr = Xraw + lane * 96;
    float s = 0.f;
    for (int c = 0; c < 96; ++c) s += xr[c];
    const float mu = s * (1.f / 96.f);
    float v = 0.f;
    for (int c = 0; c < 96; ++c) { float d = xr[c] - mu; v += d * d; }
    const float rs = rsqrtf(v * (1.f / 96.f) + 1e-5f);
    float* xo = Xln + lane * 96;
    for (int c = 0; c < 96; ++c) xo[c] = (xr[c] - mu) * rs * n1w[c] + n1b[c];
  }
  const int mrow = lane & 15;
  const int koff = (lane < 16) ? 0 : 2;
  const int prow = (lane < 16) ? 0 : 8;

  // QKV: (16x96)(96x288)
  for (int nt = 0; nt < 18; ++nt) {
    v8f acc = vzero();
    const float* wrow = qkvw + (nt * 16 + mrow) * 96;
    for (int kt = 0; kt < 24; ++kt) {
      const int k = kt * 4 + koff;
      v2f a, bf;
      a.x = Xln[mrow * 96 + k];  a.y = Xln[mrow * 96 + k + 1];
      bf.x = wrow[k];            bf.y = wrow[k + 1];
      acc = wmma_f32(a, bf, acc);
    }
    const float bv = qkvb[nt * 16 + mrow];
#pragma unroll
    for (int gi = 0; gi < 8; ++gi)
      QKV[(gi + prow) * 288 + nt * 16 + mrow] = acc[gi] + bv;
  }

  const float scale = 0.17677669529663687f;  // 1/sqrt(32)
  for (int h = 0; h < 3; ++h) {
    // scores = Q K^T
    v8f sc = vzero();
    for (int kt = 0; kt < 8; ++kt) {
      const int k = kt * 4 + koff;
      v2f a, bf;
      a.x  = QKV[mrow * 288 + h * 32 + k];
      a.y  = QKV[mrow * 288 + h * 32 + k + 1];
      bf.x = QKV[mrow * 288 + 96 + h * 32 + k];
      bf.y = QKV[mrow * 288 + 96 + h * 32 + k + 1];
      sc = wmma_f32(a, bf, sc);
    }
#pragma unroll
    for (int gi = 0; gi < 8; ++gi) Pm[(gi + prow) * 16 + mrow] = sc[gi];
    if (lane < 16) {                 // scale + rel bias + shift mask + softmax
      const int n = lane, rn = n >> 2, cn = n & 3;
      const int labn = region3(wy * 4 + rn) * 3 + region3(wx * 4 + cn);
      float rowv[16];
      float mx = -3.0e38f;
#pragma unroll
      for (int m = 0; m < 16; ++m) {
        const int rm = m >> 2, cm = m & 3;
        const int idx = (rn - rm + 3) * 7 + (cn - cm + 3);
        const int labm = region3(wy * 4 + rm) * 3 + region3(wx * 4 + cm);
        float vv = Pm[n * 16 + m] * scale + relb[idx * 3 + h] +
                   ((labn != labm) ? -100.f : 0.f);
        rowv[m] = vv;
        mx = fmaxf(mx, vv);
      }
      float ssum = 0.f;
#pragma unroll
      for (int m = 0; m < 16; ++m) { float e = expf(rowv[m] - mx); rowv[m] = e; ssum += e; }
      const float inv = 1.f / ssum;
#pragma unroll
      for (int m = 0; m < 16; ++m) Pm[n * 16 + m] = rowv[m] * inv;
    }
    // O = P V  (16x16)(16x32)
    for (int nt = 0; nt < 2; ++nt) {
      v8f ov = vzero();
      for (int kt = 0; kt < 4; ++kt) {
        const int k = kt * 4 + koff;
        v2f a, bf;
        a.x = Pm[mrow * 16 + k]; a.y = Pm[mrow * 16 + k + 1];
        bf.x = QKV[k * 288 + 192 + h * 32 + nt * 16 + mrow];
        bf.y = QKV[(k + 1) * 288 + 192 + h * 32 + nt * 16 + mrow];
        ov = wmma_f32(a, bf, ov);
      }
#pragma unroll
      for (int gi = 0; gi < 8; ++gi)
        Xln[(gi + prow) * 96 + h * 32 + nt * 16 + mrow] = ov[gi];
    }
  }

  // attn_proj (96->96) + shortcut, write back in place
  for (int nt = 0; nt < 6; ++nt) {
    v8f acc = vzero();
    const float* wrow = apw + (nt * 16 + mrow) * 96;
    for (int kt = 0; kt < 24; ++kt) {
      const int k = kt * 4 + koff;
      v2f a, bf;
      a.x = Xln[mrow * 96 + k]; a.y = Xln[mrow * 96 + k + 1];
      bf.x = wrow[k];           bf.y = wrow[k + 1];
      acc = wmma_f32(a, bf, acc);
    }
    const float bv = apb[nt * 16 + mrow];
#pragma unroll
    for (int gi = 0; gi < 8; ++gi) {
      const int n = gi + prow;
      feat[bbase + (long)win_pix(wy, wx, n) * 96 + nt * 16 + mrow] =
          Xraw[n * 96 + nt * 16 + mrow] + acc[gi] + bv;
    }
  }
}

// ---------------- kernel 5: LN + MLP(96->384->96) + residual (in place) ----
__global__ void __launch_bounds__(128) k_mlp(float* __restrict__ feat,
                                             const float* __restrict__ n2w,
                                             const float* __restrict__ n2b,
                                             const float* __restrict__ f1w,
                                             const float* __restrict__ f1b,
                                             const float* __restrict__ f2w,
                                             const float* __restrict__ f2b) {
  extern __shared__ float lds[];
  const int tid = threadIdx.x, wave = tid >> 5, lane = tid & 31;
  float* Xraw = lds + wave * 9216;   // 16x96
  float* Xln  = Xraw + 1536;         // 16x96
  float* Hid  = Xraw + 3072;         // 16x384
  const long row0 = ((long)blockIdx.x * 4 + wave) * 16;
  float* base = feat + row0 * 96;
#if HAVE_ASYNC_LDS
  for (int i = lane; i < 384; i += 32) async_b128(base + i * 4, Xraw + i * 4);
  async_wait();
#else
  for (int i = lane; i < 1536; i += 32) Xraw[i] = base[i];
#endif
  if (lane < 16) {
    const float* xr = Xraw + lane * 96;
    float s = 0.f;
    for (int c = 0; c < 96; ++c) s += xr[c];
    const float mu = s * (1.f / 96.f);
    float v = 0.f;
    for (int c = 0; c < 96; ++c) { float d = xr[c] - mu; v += d * d; }
    const float rs = rsqrtf(v * (1.f / 96.f) + 1e-5f);
    float* xo = Xln + lane * 96;
    for (int c = 0; c < 96; ++c) xo[c] = (xr[c] - mu) * rs * n2w[c] + n2b[c];
  }
  const int mrow = lane & 15;
  const int koff = (lane < 16) ? 0 : 2;
  const int prow = (lane < 16) ? 0 : 8;
  for (int nt = 0; nt < 24; ++nt) {              // fc1 + exact GELU
    v8f acc = vzero();
    const float* wrow = f1w + (nt * 16 + mrow) * 96;
    for (int kt = 0; kt < 24; ++kt) {
      const int k = kt * 4 + koff;
      v2f a, bf;
      a.x = Xln[mrow * 96 + k]; a.y = Xln[mrow * 96 + k + 1];
      bf.x = wrow[k];           bf.y = wrow[k + 1];
      acc = wmma_f32(a, bf, acc);
    }
    const float bv = f1b[nt * 16 + mrow];
#pragma unroll
    for (int gi = 0; gi < 8; ++gi) {
      float vv = acc[gi] + bv;
      vv = 0.5f * vv * (1.f + erff(vv * 0.70710678118654752f));
      Hid[(gi + prow) * 384 + nt * 16 + mrow] = vv;
    }
  }
  for (int nt = 0; nt < 6; ++nt) {               // fc2 + residual
    v8f acc = vzero();
    const float* wrow = f2w + (nt * 16 + mrow) * 384;
    for (int kt = 0; kt < 96; ++kt) {
      const int k = kt * 4 + koff;
      v2f a, bf;
      a.x = Hid[mrow * 384 + k]; a.y = Hid[mrow * 384 + k + 1];
      bf.x = wrow[k];            bf.y = wrow[k + 1];
      acc = wmma_f32(a, bf, acc);
    }
    const float bv = f2b[nt * 16 + mrow];
#pragma unroll
    for (int gi = 0; gi < 8; ++gi)
      base[(gi + prow) * 96 + nt * 16 + mrow] =
          Xraw[(gi + prow) * 96 + nt * 16 + mrow] + acc[gi] + bv;
  }
}

// ------- kernel 6: proj_out (96->128) + gated residual into NCHW out -------
__global__ void __launch_bounds__(128) k_proj_out(const float* __restrict__ feat,
                                                  const float* __restrict__ pw,
                                                  const float* __restrict__ pb,
                                                  const float* __restrict__ x,
                                                  const float* __restrict__ gate,
                                                  float* __restrict__ out) {
  extern __shared__ float lds[];
  const int tid = threadIdx.x, wave = tid >> 5, lane = tid & 31;
  float* Xt = lds + wave * 3584;     // 16x96 input tile
  float* T  = Xt + 1536;             // 128x16 transposed result
  const long row0 = ((long)blockIdx.x * 4 + wave) * 16;
  const int b  = (int)(row0 / HW);
  const int p0 = (int)(row0 - (long)b * HW);
  const float* src = feat + row0 * 96;
#if HAVE_ASYNC_LDS
  for (int i = lane; i < 384; i += 32) async_b128(src + i * 4, Xt + i * 4);
  async_wait();
#else
  for (int i = lane; i < 1536; i += 32) Xt[i] = src[i];
#endif
  const int mrow = lane & 15;
  const int koff = (lane < 16) ? 0 : 2;
  const int prow = (lane < 16) ? 0 : 8;
  for (int nt = 0; nt < 8; ++nt) {
    v8f acc = vzero();
    const float* wrow = pw + (nt * 16 + mrow) * 96;
    for (int kt = 0; kt < 24; ++kt) {
      const int k = kt * 4 + koff;
      v2f a, bf;
      a.x = Xt[mrow * 96 + k]; a.y = Xt[mrow * 96 + k + 1];
      bf.x = wrow[k];          bf.y = wrow[k + 1];
      acc = wmma_f32(a, bf, acc);
    }
    const float bv = pb[nt * 16 + mrow];
#pragma unroll
    for (int gi = 0; gi < 8; ++gi)
      T[(nt * 16 + mrow) * 16 + gi + prow] = acc[gi] + bv;
  }
  // coalesced float4 NT stores into (B,128,H,W) with x + t*g fusion
  for (int i = lane; i < 512; i += 32) {
    const int o = i >> 2, ch = i & 3;
    const long addr = ((long)b * 128 + o) * HW + p0 + ch * 4;
    const v4f xv = __builtin_nontemporal_load((const v4f*)(x + addr));
    const float gv = gate[b * 128 + o];
    const v4f tv = *(const v4f*)(T + o * 16 + ch * 4);
    v4f r;
    r.x = xv.x + tv.x * gv;
    r.y = xv.y + tv.y * gv;
    r.z = xv.z + tv.z * gv;
    r.w = xv.w + tv.w * gv;
    __builtin_nontemporal_store(r, (v4f*)(out + addr));
  }
}

extern "C" void kernel_launch(void* const* d_in, const int* in_sizes, int n_in,
                              void* d_out, int out_size, void* d_ws, size_t ws_size,
                              hipStream_t stream) {
  (void)in_sizes; (void)n_in; (void)out_size; (void)ws_size;
  const float* x    = (const float*)d_in[0];
  const float* piw  = (const float*)d_in[1];
  const float* pib  = (const float*)d_in[2];
  const float* n1w  = (const float*)d_in[3];
  const float* n1b  = (const float*)d_in[4];
  const float* qkvw = (const float*)d_in[5];
  const float* qkvb = (const float*)d_in[6];
  const float* relb = (const float*)d_in[7];
  const float* apw  = (const float*)d_in[8];
  const float* apb  = (const float*)d_in[9];
  const float* n2w  = (const float*)d_in[10];
  const float* n2b  = (const float*)d_in[11];
  const float* f1w  = (const float*)d_in[12];
  const float* f1b  = (const float*)d_in[13];
  const float* f2w  = (const float*)d_in[14];
  const float* f2b  = (const float*)d_in[15];
  const float* pow_ = (const float*)d_in[16];
  const float* pob  = (const float*)d_in[17];
  const float* w1   = (const float*)d_in[18];
  const float* b1   = (const float*)d_in[19];
  const float* w2   = (const float*)d_in[20];
  const float* b2   = (const float*)d_in[21];
  float* out = (float*)d_out;

  float* ws   = (float*)d_ws;
  float* g0   = ws;            // 1024
  float* gacc = ws + 1024;     // 1024
  float* feat = ws + 2048;     // 8*50176*96 floats (~154 MB), reused in place

  k_mean    <<<1024, 256, 0,      stream>>>(x, g0);
  k_gate    <<<8,    128, 0,      stream>>>(g0, w1, b1, w2, b2, gacc);
  k_proj_in <<<6272, 128, 49152,  stream>>>(x, piw, pib, feat);
  k_attn    <<<6272, 128, 126976, stream>>>(feat, n1w, n1b, qkvw, qkvb, relb, apw, apb);
  k_mlp     <<<6272, 128, 147456, stream>>>(feat, n2w, n2b, f1w, f1b, f2w, f2b);
  k_proj_out<<<6272, 128, 57344,  stream>>>(feat, pow_, pob, x, gacc, out);
}